// BayesLinearMF_37074157699683
// MI455X (gfx1250) — compile-verified
//
#include <hip/hip_runtime.h>

typedef float v2f __attribute__((ext_vector_type(2)));
typedef float v4f __attribute__((ext_vector_type(4)));
typedef float v8f __attribute__((ext_vector_type(8)));

#define BS  128
#define IN  1024
#define OUT 1024

// ---------------------------------------------------------------------------
// Kernel A: out[b,o] = sum_i x[b,i]*mu[o,i] + bias_mu[o] + eps_b[b,o]*exp(bias_psi[o])
// One wave32 per 16x16 (b,o) tile using V_WMMA_F32_16X16X4_F32, K swept 4 at a time.
// A-matrix (16x4 f32) layout per ISA: lanes 0-15 hold M=lane; VGPR0 = K=0 (lo half)/K=2 (hi half),
// VGPR1 = K=1/K=3.  B (4x16) mirrors with N=lane.  C/D: VGPR r -> M=r(+8 for hi half), N=lane.
// ---------------------------------------------------------------------------
__global__ __launch_bounds__(32) void bayes_mu_gemm_wmma(
    const float* __restrict__ x,        // (BS, IN)
    const float* __restrict__ mu,       // (OUT, IN)
    const float* __restrict__ bias_mu,  // (OUT)
    const float* __restrict__ bias_psi, // (OUT)
    const float* __restrict__ eps_b,    // (BS, OUT)
    float* __restrict__ out)            // (BS, OUT)
{
    const int tile = blockIdx.x;         // 0..511
    const int bt   = tile >> 6;          // 8 b-tiles
    const int ot   = tile & 63;          // 64 o-tiles
    const int b0   = bt << 4;
    const int o0   = ot << 4;

    const int lane = threadIdx.x & 31;
    const int m    = lane & 15;
    const int hi   = lane >> 4;          // 0 or 1 -> K offset 0 or 2

    const float* __restrict__ xrow  = x  + (size_t)(b0 + m) * IN;  // A: M = m
    const float* __restrict__ murow = mu + (size_t)(o0 + m) * IN;  // B: N = m

    v8f c = {};
    for (int k0 = 0; k0 < IN; k0 += 4) {
        const int k = k0 + 2 * hi;
        v2f a, b;
        a.x = xrow[k];
        a.y = xrow[k + 1];
        b.x = murow[k];
        b.y = murow[k + 1];
        c = __builtin_amdgcn_wmma_f32_16x16x4_f32(
                /*neg_a=*/false, a, /*neg_b=*/false, b,
                /*c_mod=*/(short)0, c, /*reuse_a=*/false, /*reuse_b=*/false);
    }

    // epilogue: add reparameterized bias, store D
    const int   o   = o0 + m;
    const float ebp = __expf(bias_psi[o]);
    const float bmu = bias_mu[o];
#pragma unroll
    for (int r = 0; r < 8; ++r) {
        const int b = b0 + r + 8 * hi;
        const size_t idx = (size_t)b * OUT + o;
        out[idx] = c[r] + fmaf(eps_b[idx], ebp, bmu);
    }
}

// ---------------------------------------------------------------------------
// Kernel B: out[b,o] += sum_i eps_w[b,o,i] * exp(weight_psi[o,i]) * x[b,i]
// One block per output row o.  s[i] = exp(psi[o,i]) computed once into LDS and
// reused across all 128 batches (amortizes the transcendentals 128x).
// eps_w is a 512MB single-use stream -> non-temporal b128 loads keep L2 clean.
// Each of the 8 waves owns batches {wave, wave+8, ...}; wave32 xor-shuffle reduce.
// ---------------------------------------------------------------------------
__global__ __launch_bounds__(256) void bayes_eps_stream(
    const float* __restrict__ x,      // (BS, IN)
    const float* __restrict__ psi,    // (OUT, IN)
    const float* __restrict__ eps_w,  // (BS, OUT, IN)
    float* __restrict__ out)          // (BS, OUT)
{
    __shared__ float s_s[IN];

    const int o   = blockIdx.x;
    const int tid = threadIdx.x;

    // build s row: 256 threads * 4 elements
    {
        const v4f p = *(const v4f*)(psi + (size_t)o * IN + tid * 4);
        v4f e;
        e.x = __expf(p.x);
        e.y = __expf(p.y);
        e.z = __expf(p.z);
        e.w = __expf(p.w);
        *(v4f*)(s_s + tid * 4) = e;
    }
    __syncthreads();

    const int wave = tid >> 5;
    const int lane = tid & 31;

    for (int b = wave; b < BS; b += 8) {
        const float* __restrict__ erow = eps_w + ((size_t)b * OUT + o) * IN;
        const float* __restrict__ xrow = x + (size_t)b * IN;

        float acc = 0.0f;
#pragma unroll
        for (int ii = 0; ii < IN; ii += 128) {
            const int i = ii + lane * 4;
            const v4f e  = __builtin_nontemporal_load((const v4f*)(erow + i));
            const v4f xv = *(const v4f*)(xrow + i);
            const v4f sv = *(const v4f*)(s_s + i);
            acc = fmaf(e.x, sv.x * xv.x, acc);
            acc = fmaf(e.y, sv.y * xv.y, acc);
            acc = fmaf(e.z, sv.z * xv.z, acc);
            acc = fmaf(e.w, sv.w * xv.w, acc);
        }

        // wave32 reduction
#pragma unroll
        for (int off = 16; off >= 1; off >>= 1)
            acc += __shfl_xor(acc, off, 32);

        if (lane == 0)
            out[(size_t)b * OUT + o] += acc;   // kernel A already wrote mu+bias term
    }
}

extern "C" void kernel_launch(void* const* d_in, const int* in_sizes, int n_in,
                              void* d_out, int out_size, void* d_ws, size_t ws_size,
                              hipStream_t stream) {
    (void)in_sizes; (void)n_in; (void)d_ws; (void)ws_size; (void)out_size;

    const float* x    = (const float*)d_in[0];
    const float* wmu  = (const float*)d_in[1];
    const float* wpsi = (const float*)d_in[2];
    const float* bmu  = (const float*)d_in[3];
    const float* bpsi = (const float*)d_in[4];
    const float* epsw = (const float*)d_in[5];
    const float* epsb = (const float*)d_in[6];
    float* out = (float*)d_out;

    // mu-GEMM + bias (WMMA), then the bandwidth-bound eps stream accumulates on top.
    bayes_mu_gemm_wmma<<<dim3(512), dim3(32), 0, stream>>>(x, wmu, bmu, bpsi, epsb, out);
    bayes_eps_stream<<<dim3(OUT), dim3(256), 0, stream>>>(x, wpsi, epsw, out);
}